// MultiHeadedAttention_30210799960138
// MI455X (gfx1250) — compile-verified
//
#include <hip/hip_runtime.h>

// ---------------------------------------------------------------------------
// MI455X (gfx1250) fp32 multi-head attention, all matmuls on the WMMA pipe
// via V_WMMA_F32_16X16X4_F32 (exact fp32 numerics vs the fp32 reference).
// B=2, T=2048, E=2048, H=8, KV=1, D=256.
// ---------------------------------------------------------------------------

typedef __attribute__((ext_vector_type(2))) float v2f;
typedef __attribute__((ext_vector_type(8))) float v8f;

__device__ __forceinline__ v8f wmma_f32(v2f a, v2f b, v8f c) {
  // 8-arg form: (neg_a, A, neg_b, B, c_mod, C, reuse_a, reuse_b)
  return __builtin_amdgcn_wmma_f32_16x16x4_f32(false, a, false, b, (short)0, c,
                                               false, false);
}

// ------------------------------- GEMM --------------------------------------
// C[M,N] = A[M,K] * B[K,N] + bias[N]; 256 threads = 8 waves.
// Block tile 128x128, BK=16; wave tile 32x64 = 2x4 grid of 16x16 fragments.
#define GAS 20   // A LDS row stride (128 x 16 tile), conflict-free frag reads
#define GBS 136  // B LDS row stride (16 x 128 tile), 2*GBS % 64 == 16

__global__ __launch_bounds__(256) void gemm_bias_kernel(
    const float* __restrict__ A, const float* __restrict__ B,
    const float* __restrict__ bias, float* __restrict__ C, int M, int N,
    int K) {
  extern __shared__ float lds[];
  float* As = lds;              // [128][GAS]
  float* Bs = lds + 128 * GAS;  // [16][GBS]

  const int tid = threadIdx.x;
  const int lane = tid & 31;
  const int ln = lane & 15;
  const int half = lane >> 4;
  const int w = tid >> 5;
  const int wrow = w >> 1;  // 0..3 : 32-row slice
  const int wcol = w & 1;   // 0..1 : 64-col slice

  const int m0 = blockIdx.x * 128;
  const int n0 = blockIdx.y * 128;

  const int ar = tid >> 1;        // 0..127
  const int ac = (tid & 1) * 8;   // 0 or 8
  const int br = tid >> 4;        // 0..15
  const int bc = (tid & 15) * 8;  // 0..120

  v8f acc[2][4];
#pragma unroll
  for (int ms = 0; ms < 2; ++ms)
#pragma unroll
    for (int ns = 0; ns < 4; ++ns)
#pragma unroll
      for (int q = 0; q < 8; ++q) acc[ms][ns][q] = 0.0f;

  float4 a0, a1, b0, b1;
  {
    const float* ap = A + (size_t)(m0 + ar) * K + ac;
    a0 = *(const float4*)(ap);
    a1 = *(const float4*)(ap + 4);
    const float* bp = B + (size_t)br * N + n0 + bc;
    b0 = *(const float4*)(bp);
    b1 = *(const float4*)(bp + 4);
  }

  const int nk = K >> 4;
  for (int t = 0; t < nk; ++t) {
    __syncthreads();
    *(float4*)(As + ar * GAS + ac) = a0;
    *(float4*)(As + ar * GAS + ac + 4) = a1;
    *(float4*)(Bs + br * GBS + bc) = b0;
    *(float4*)(Bs + br * GBS + bc + 4) = b1;
    __syncthreads();
    if (t + 1 < nk) {  // register prefetch of next k-slice overlaps WMMA work
      const int kk = (t + 1) << 4;
      const float* ap = A + (size_t)(m0 + ar) * K + kk + ac;
      a0 = *(const float4*)(ap);
      a1 = *(const float4*)(ap + 4);
      const float* bp = B + (size_t)(kk + br) * N + n0 + bc;
      b0 = *(const float4*)(bp);
      b1 = *(const float4*)(bp + 4);
    }
#pragma unroll
    for (int kc = 0; kc < 16; kc += 4) {
      v2f afrag[2];
#pragma unroll
      for (int ms = 0; ms < 2; ++ms) {
        const int m = wrow * 32 + ms * 16 + ln;
        afrag[ms].x = As[m * GAS + kc + 2 * half];
        afrag[ms].y = As[m * GAS + kc + 2 * half + 1];
      }
      v2f bfrag[4];
#pragma unroll
      for (int ns = 0; ns < 4; ++ns) {
        const int n = wcol * 64 + ns * 16 + ln;
        bfrag[ns].x = Bs[(kc + 2 * half) * GBS + n];
        bfrag[ns].y = Bs[(kc + 2 * half + 1) * GBS + n];
      }
#pragma unroll
      for (int ms = 0; ms < 2; ++ms)
#pragma unroll
        for (int ns = 0; ns < 4; ++ns)
          acc[ms][ns] = wmma_f32(afrag[ms], bfrag[ns], acc[ms][ns]);
    }
  }

#pragma unroll
  for (int ns = 0; ns < 4; ++ns) {
    const int n = n0 + wcol * 64 + ns * 16 + ln;
    const float bv = bias[n];
#pragma unroll
    for (int ms = 0; ms < 2; ++ms) {
#pragma unroll
      for (int j = 0; j < 8; ++j) {
        const int row = m0 + wrow * 32 + ms * 16 + half * 8 + j;
        C[(size_t)row * N + n] = acc[ms][ns][j] + bv;
      }
    }
  }
}

// ----------------------------- Attention -----------------------------------
// Flash-style online softmax. Block = 128 threads (4 waves), 64 query rows of
// one (b,h); each wave owns 16 rows. 32 key tiles of 64.
#define AQS 260  // Q LDS row stride  (64 x 256)
#define AKS 72   // K^T LDS row stride (256 x 64), 2*AKS % 64 == 16
#define AVS 264  // V LDS row stride  (64 x 256),  2*AVS % 64 == 16
#define APS 68   // P LDS row stride  (16 x 64 per wave)
#define T_LEN 2048
#define HD 256

__global__ __launch_bounds__(128) void attention_kernel(
    const float* __restrict__ Q, const float* __restrict__ Kb,
    const float* __restrict__ Vb, const float* __restrict__ mask,
    float* __restrict__ O2) {
  extern __shared__ float lds[];
  float* Qs = lds;             // [64][AQS]
  float* Ks = Qs + 64 * AQS;   // [256][AKS] transposed tile: (d, key)
  float* Vs = Ks + 256 * AKS;  // [64][AVS]
  float* Ps = Vs + 64 * AVS;   // [4][16][APS] wave-private P staging

  const int tid = threadIdx.x;
  const int lane = tid & 31;
  const int ln = lane & 15;
  const int half = lane >> 4;
  const int w = tid >> 5;

  const int qt = blockIdx.x;  // 0..31
  const int bh = blockIdx.y;  // 0..15  (= b*8 + h, matching the reshape view)
  const int b = bh >> 3;
  const int h = bh & 7;
  const int qbase = qt * 64;

  const float* Qp = Q + ((size_t)bh * T_LEN + qbase) * HD;
  const float* Kp = Kb + (size_t)b * T_LEN * HD;  // KV head shared across h
  const float* Vp = Vb + (size_t)b * T_LEN * HD;
  const float* Mp = mask + (size_t)b * T_LEN * T_LEN;

  // Q tile resident for all 32 key tiles
  for (int i = tid; i < 64 * (HD / 4); i += 128) {
    const int r = i >> 6;
    const int c = (i & 63) * 4;
    *(float4*)(Qs + r * AQS + c) = *(const float4*)(Qp + (size_t)r * HD + c);
  }

  v8f O[16];
#pragma unroll
  for (int dt = 0; dt < 16; ++dt)
#pragma unroll
    for (int j = 0; j < 8; ++j) O[dt][j] = 0.0f;
  float mrow[8], lrow[8];
#pragma unroll
  for (int j = 0; j < 8; ++j) {
    mrow[j] = -3.0e38f;
    lrow[j] = 0.0f;
  }

  for (int kt = 0; kt < T_LEN / 64; ++kt) {
    const int kb0 = kt * 64;
    __syncthreads();  // previous tile's compute done with Ks/Vs
    for (int i = tid; i < 64 * (HD / 4); i += 128) {  // K tile, transposed
      const int key = i >> 6;
      const int c = (i & 63) * 4;
      const float4 kv = *(const float4*)(Kp + (size_t)(kb0 + key) * HD + c);
      Ks[(c + 0) * AKS + key] = kv.x;
      Ks[(c + 1) * AKS + key] = kv.y;
      Ks[(c + 2) * AKS + key] = kv.z;
      Ks[(c + 3) * AKS + key] = kv.w;
    }
    for (int i = tid; i < 64 * (HD / 4); i += 128) {  // V tile
      const int r = i >> 6;
      const int c = (i & 63) * 4;
      *(float4*)(Vs + r * AVS + c) =
          *(const float4*)(Vp + (size_t)(kb0 + r) * HD + c);
    }
    __syncthreads();

    // S = Q * K^T : 16 rows x 64 keys per wave, contraction D=256
    v8f S[4];
#pragma unroll
    for (int nt = 0; nt < 4; ++nt)
#pragma unroll
      for (int j = 0; j < 8; ++j) S[nt][j] = 0.0f;
#pragma unroll 4
    for (int k0 = 0; k0 < HD; k0 += 4) {
      v2f a;
      const int qrow = w * 16 + ln;
      a.x = Qs[qrow * AQS + k0 + 2 * half];
      a.y = Qs[qrow * AQS + k0 + 2 * half + 1];
#pragma unroll
      for (int nt = 0; nt < 4; ++nt) {
        v2f bb;
        const int col = nt * 16 + ln;
        bb.x = Ks[(k0 + 2 * half) * AKS + col];
        bb.y = Ks[(k0 + 2 * half + 1) * AKS + col];
        S[nt] = wmma_f32(a, bb, S[nt]);
      }
    }

    // scale (* sqrt(D) = 16, faithful quirk) + mask + online softmax update
    float corr[8];
#pragma unroll
    for (int j = 0; j < 8; ++j) {
      const int qrow = qbase + w * 16 + half * 8 + j;
      float mx = -3.0e38f;
#pragma unroll
      for (int nt = 0; nt < 4; ++nt) {
        const int kcol = kb0 + nt * 16 + ln;
        const float sv = S[nt][j] * 16.0f + Mp[(size_t)qrow * T_LEN + kcol];
        S[nt][j] = sv;
        mx = fmaxf(mx, sv);
      }
#pragma unroll
      for (int off = 1; off < 16; off <<= 1)
        mx = fmaxf(mx, __shfl_xor(mx, off, 32));
      const float mnew = fmaxf(mrow[j], mx);
      corr[j] = __expf(mrow[j] - mnew);
      mrow[j] = mnew;
      float psum = 0.0f;
#pragma unroll
      for (int nt = 0; nt < 4; ++nt) {
        const float p = __expf(S[nt][j] - mnew);
        S[nt][j] = p;
        psum += p;
      }
#pragma unroll
      for (int off = 1; off < 16; off <<= 1)
        psum += __shfl_xor(psum, off, 32);
      lrow[j] = lrow[j] * corr[j] + psum;
    }

    // rescale O; spill P (C-layout -> A-layout) to wave-private LDS
    float* Pw = Ps + w * 16 * APS;
#pragma unroll
    for (int j = 0; j < 8; ++j) {
#pragma unroll
      for (int dt = 0; dt < 16; ++dt) O[dt][j] *= corr[j];
      const int prow = half * 8 + j;
#pragma unroll
      for (int nt = 0; nt < 4; ++nt)
        Pw[prow * APS + nt * 16 + ln] = S[nt][j];
    }

    // O += P * V : 16 rows x 256 dims, contraction over 64 keys
#pragma unroll 2
    for (int k0 = 0; k0 < 64; k0 += 4) {
      v2f a;
      a.x = Pw[ln * APS + k0 + 2 * half];
      a.y = Pw[ln * APS + k0 + 2 * half + 1];
#pragma unroll
      for (int dt = 0; dt < 16; ++dt) {
        v2f bb;
        const int col = dt * 16 + ln;
        bb.x = Vs[(k0 + 2 * half) * AVS + col];
        bb.y = Vs[(k0 + 2 * half + 1) * AVS + col];
        O[dt] = wmma_f32(a, bb, O[dt]);
      }
    }
  }

  // normalize; store directly in (b, t, h*HD + d) so Wo GEMM is plain
  float rinv[8];
#pragma unroll
  for (int j = 0; j < 8; ++j) rinv[j] = 1.0f / lrow[j];
#pragma unroll
  for (int dt = 0; dt < 16; ++dt) {
#pragma unroll
    for (int j = 0; j < 8; ++j) {
      const int t2 = qbase + w * 16 + half * 8 + j;
      const int col = h * HD + dt * 16 + ln;
      O2[((size_t)b * T_LEN + t2) * 2048 + col] = O[dt][j] * rinv[j];
    }
  }
}

// ------------------------------- Launch ------------------------------------
extern "C" void kernel_launch(void* const* d_in, const int* in_sizes, int n_in,
                              void* d_out, int out_size, void* d_ws,
                              size_t ws_size, hipStream_t stream) {
  (void)in_sizes;
  (void)n_in;
  (void)out_size;
  (void)ws_size;
  const float* x = (const float*)d_in[0];
  const float* mask = (const float*)d_in[1];
  const float* Wq = (const float*)d_in[2];
  const float* bq = (const float*)d_in[3];
  const float* Wk = (const float*)d_in[4];
  const float* bk = (const float*)d_in[5];
  const float* Wv = (const float*)d_in[6];
  const float* bv = (const float*)d_in[7];
  const float* Wo = (const float*)d_in[8];
  const float* bo = (const float*)d_in[9];
  float* out = (float*)d_out;

  // workspace: Q (B*T*E) | K (B*T*256) | V (B*T*256) | O2 (B*T*E)
  float* ws = (float*)d_ws;
  float* Qb = ws;
  float* Kb = Qb + (size_t)8388608;
  float* Vb = Kb + (size_t)1048576;
  float* O2 = Vb + (size_t)1048576;

  const int M = 4096, E = 2048, KV = 256;
  const size_t gemmLds = (size_t)(128 * GAS + 16 * GBS) * sizeof(float);
  const size_t attnLds =
      (size_t)(64 * AQS + 256 * AKS + 64 * AVS + 4 * 16 * APS) * sizeof(float);

  gemm_bias_kernel<<<dim3(M / 128, E / 128), 256, gemmLds, stream>>>(
      x, Wq, bq, Qb, M, E, E);
  gemm_bias_kernel<<<dim3(M / 128, KV / 128), 256, gemmLds, stream>>>(
      x, Wk, bk, Kb, M, KV, E);
  gemm_bias_kernel<<<dim3(M / 128, KV / 128), 256, gemmLds, stream>>>(
      x, Wv, bv, Vb, M, KV, E);
  attention_kernel<<<dim3(32, 16), 128, attnLds, stream>>>(Qb, Kb, Vb, mask,
                                                           O2);
  gemm_bias_kernel<<<dim3(M / 128, E / 128), 256, gemmLds, stream>>>(
      O2, Wo, bo, out, M, E, E);
}